// Model_11879879543299
// MI455X (gfx1250) — compile-verified
//
#include <hip/hip_runtime.h>
#include <hip/hip_bf16.h>

// Problem constants (match reference)
#define TT 256
#define BB 64
#define DD 256
#define UU 512
#define OO 10
#define LAMBC 0.01f
#define APERT 0.0625f               /* 4^-2 */
#define GAMC (1.0f - LAMBC*APERT)   /* decay of C per step */
#define EPSLN 1e-6f

typedef float v2f __attribute__((ext_vector_type(2)));
typedef float v8f __attribute__((ext_vector_type(8)));

// ---------------------------------------------------------------------------
// Kernel 1: xk[t][b][u] = sum_d x[b][t][d] * K[d][u] + bias[u]
// One wave per 16(b) x 64(u) macro-tile: the A fragment (b64 load) is reused
// across 4 V_WMMA_F32_16X16X4_F32 issues per K-step.
// A (16x4) per-lane: {A[m][kk], A[m][kk+1]}, m=lane%16, kk=(lane/16)*2
// B (4x16) per-lane: {B[kk][n],  B[kk+1][n]}, n=lane%16
// C/D (16x16): VGPR r holds M = r + 8*(lane/16), N = lane%16
// ---------------------------------------------------------------------------
__global__ __launch_bounds__(256) void xk_gemm_kernel(const float* __restrict__ x,
                                                      const float* __restrict__ kmat,
                                                      const float* __restrict__ bias,
                                                      float* __restrict__ xk)
{
    const int wid  = blockIdx.x * 8 + (threadIdx.x >> 5);   // global wave id
    const int lane = threadIdx.x & 31;
    const int t    = wid >> 5;            // 32 macro-tiles per t (4 b * 8 u-quads)
    const int rem  = wid & 31;
    const int b0   = (rem >> 3) << 4;     // batch tile base
    const int u0   = (rem & 7) << 6;      // u macro-tile base (64 wide)
    const int m    = lane & 15;
    const int n    = lane & 15;
    const int kk   = (lane >> 4) << 1;    // K sub-offset for this half-wave

    const float* arow = x + (size_t)(b0 + m) * (TT * DD) + (size_t)t * DD;
    const float* bcol = kmat + (u0 + n);

    v8f c0 = {}, c1 = {}, c2 = {}, c3 = {};
    #pragma unroll 4
    for (int k0 = 0; k0 < DD; k0 += 4) {
        v2f a;
        a.x = arow[k0 + kk];
        a.y = arow[k0 + kk + 1];
        const float* brow0 = bcol + (size_t)(k0 + kk) * UU;
        const float* brow1 = bcol + (size_t)(k0 + kk + 1) * UU;
        v2f bv0; bv0.x = brow0[0];  bv0.y = brow1[0];
        v2f bv1; bv1.x = brow0[16]; bv1.y = brow1[16];
        v2f bv2; bv2.x = brow0[32]; bv2.y = brow1[32];
        v2f bv3; bv3.x = brow0[48]; bv3.y = brow1[48];
        // (neg_a, A, neg_b, B, c_mod, C, reuse_a, reuse_b)
        c0 = __builtin_amdgcn_wmma_f32_16x16x4_f32(false, a, false, bv0, (short)0, c0, false, false);
        c1 = __builtin_amdgcn_wmma_f32_16x16x4_f32(false, a, false, bv1, (short)0, c1, false, false);
        c2 = __builtin_amdgcn_wmma_f32_16x16x4_f32(false, a, false, bv2, (short)0, c2, false, false);
        c3 = __builtin_amdgcn_wmma_f32_16x16x4_f32(false, a, false, bv3, (short)0, c3, false, false);
    }

    const int  mbase = (lane >> 4) << 3;   // 0 or 8
    float* obase = xk + (size_t)t * (BB * UU);
    #pragma unroll
    for (int j = 0; j < 4; ++j) {
        const int   u  = u0 + 16 * j + n;
        const float bu = bias[u];
        const v8f&  c  = (j == 0) ? c0 : (j == 1) ? c1 : (j == 2) ? c2 : c3;
        float* orow = obase + u;
        #pragma unroll
        for (int r = 0; r < 8; ++r) {
            orow[(size_t)(b0 + mbase + r) * UU] = c[r] + bu;
        }
    }
}

// ---------------------------------------------------------------------------
// Kernel 2: persistent scan. One workgroup (1024 thr = 32 waves) per batch b.
// C_b (512x512 f32, 1 MiB) stays L2-resident; every element is owned by one
// thread (tid -> rows [ug*64,+64), cols [v4,+4)): C coherence is same-thread.
// The rank-1 update of step t-1 is applied LAZILY, fused into step t's hC
// read pass -> exactly one load + one store of each C element per step:
//   A : hcur = tanh(xk[t,b,:] + hprev @ R)
//   B': c = GAM*c + coef[u]*hold[v]; store c; hC[v] += hcur[u]*c   (fused)
//   C : hh = h.h, hhc = h.hC (wave-0 shuffle reduce)
//   D : y = GAM*hC + LAMB*(hh-hhc)*h ; hprev = tanh(LayerNorm(y));
//       coef = LAMB*(h - hC); hold = h      (pending update for step t+1)
// Final: logits = h @ dense_w + dense_b ; softmax -> out[b]
// ---------------------------------------------------------------------------
__global__ __launch_bounds__(1024) void scan_kernel(const float* __restrict__ xk,
                                                    const float* __restrict__ R,
                                                    const float* __restrict__ lng,
                                                    const float* __restrict__ lnb,
                                                    const float* __restrict__ dw,
                                                    const float* __restrict__ db,
                                                    float* __restrict__ Cg,
                                                    float* __restrict__ out)
{
    __shared__ float sh_prev[UU];
    __shared__ float sh_cur[UU];
    __shared__ float sh_hC[UU];
    __shared__ float sh_y[UU];
    __shared__ float sh_coef[UU];
    __shared__ float sh_hold[UU];
    __shared__ float sp[8 * UU];     // matvec partials / reduction scratch
    __shared__ float sred[2];

    const int tid = threadIdx.x;
    const int b   = blockIdx.x;
    const int v4  = (tid & 127) << 2;   // owned column base (4 cols)
    const int ug  = tid >> 7;           // owned row group (64 rows)

    float* Cb   = Cg + (size_t)b * UU * UU;
    float* Cown = Cb + (size_t)(ug * 64) * UU + v4;

    // Zero owned slice of C (deterministic init each call; same-thread mapping)
    for (int w = 0; w < 64; ++w) {
        float4 z = {0.f, 0.f, 0.f, 0.f};
        *(float4*)(Cown + (size_t)w * UU) = z;
    }
    if (tid < UU) { sh_prev[tid] = 0.f; sh_coef[tid] = 0.f; sh_hold[tid] = 0.f; }
    __syncthreads();

    for (int t = 0; t < TT; ++t) {
        // ---- Phase A: recurrent matvec partials ----
        {
            const float* Rp = R + (size_t)(ug * 64) * UU + v4;
            float ax = 0.f, ay = 0.f, az = 0.f, aw = 0.f;
            #pragma unroll 4
            for (int w = 0; w < 64; ++w) {
                const float hw = sh_prev[ug * 64 + w];
                const float4 r4 = *(const float4*)(Rp + (size_t)w * UU);
                ax = fmaf(hw, r4.x, ax); ay = fmaf(hw, r4.y, ay);
                az = fmaf(hw, r4.z, az); aw = fmaf(hw, r4.w, aw);
            }
            sp[ug * UU + v4 + 0] = ax;  sp[ug * UU + v4 + 1] = ay;
            sp[ug * UU + v4 + 2] = az;  sp[ug * UU + v4 + 3] = aw;
        }
        __syncthreads();
        if (tid < UU) {
            float s = xk[(size_t)t * (BB * UU) + (size_t)b * UU + tid];
            #pragma unroll
            for (int g = 0; g < 8; ++g) s += sp[g * UU + tid];
            sh_cur[tid] = tanhf(s);
        }
        __syncthreads();

        // ---- Phase B': apply pending update (lazy) + accumulate hC ----
        {
            const float hold0 = sh_hold[v4 + 0], hold1 = sh_hold[v4 + 1];
            const float hold2 = sh_hold[v4 + 2], hold3 = sh_hold[v4 + 3];
            float ax = 0.f, ay = 0.f, az = 0.f, aw = 0.f;
            #pragma unroll 4
            for (int w = 0; w < 64; ++w) {
                const int   u  = ug * 64 + w;
                const float cf = sh_coef[u];
                float4 c4 = *(float4*)(Cown + (size_t)w * UU);
                c4.x = fmaf(cf, hold0, GAMC * c4.x);
                c4.y = fmaf(cf, hold1, GAMC * c4.y);
                c4.z = fmaf(cf, hold2, GAMC * c4.z);
                c4.w = fmaf(cf, hold3, GAMC * c4.w);
                *(float4*)(Cown + (size_t)w * UU) = c4;
                const float hw = sh_cur[u];
                ax = fmaf(hw, c4.x, ax); ay = fmaf(hw, c4.y, ay);
                az = fmaf(hw, c4.z, az); aw = fmaf(hw, c4.w, aw);
            }
            sp[ug * UU + v4 + 0] = ax;  sp[ug * UU + v4 + 1] = ay;
            sp[ug * UU + v4 + 2] = az;  sp[ug * UU + v4 + 3] = aw;
        }
        __syncthreads();
        if (tid < UU) {
            float s = 0.f;
            #pragma unroll
            for (int g = 0; g < 8; ++g) s += sp[g * UU + tid];
            sh_hC[tid] = s;
            const float h = sh_cur[tid];
            sp[tid]      = h * h;     // safe: sp[tid]/sp[UU+tid] only read by this tid
            sp[UU + tid] = h * s;
        }
        __syncthreads();
        if (tid < 32) {
            float s1 = 0.f, s2 = 0.f;
            for (int i = tid; i < UU; i += 32) { s1 += sp[i]; s2 += sp[UU + i]; }
            #pragma unroll
            for (int o = 16; o > 0; o >>= 1) {
                s1 += __shfl_xor(s1, o, 32);
                s2 += __shfl_xor(s2, o, 32);
            }
            if (tid == 0) { sred[0] = s1; sred[1] = s2; }
        }
        __syncthreads();
        const float hh  = sred[0];
        const float hhc = sred[1];

        // ---- Phase D: y = GAM*hC + LAMB*(hh-hhc)*h ; stats for LN ----
        if (tid < UU) {
            const float y = fmaf(LAMBC * (hh - hhc), sh_cur[tid], GAMC * sh_hC[tid]);
            sh_y[tid]    = y;
            sp[tid]      = y;
            sp[UU + tid] = y * y;
        }
        __syncthreads();
        if (tid < 32) {
            float s1 = 0.f, s2 = 0.f;
            for (int i = tid; i < UU; i += 32) { s1 += sp[i]; s2 += sp[UU + i]; }
            #pragma unroll
            for (int o = 16; o > 0; o >>= 1) {
                s1 += __shfl_xor(s1, o, 32);
                s2 += __shfl_xor(s2, o, 32);
            }
            if (tid == 0) { sred[0] = s1; sred[1] = s2; }
        }
        __syncthreads();
        {
            const float mu   = sred[0] * (1.0f / UU);
            const float var  = sred[1] * (1.0f / UU) - mu * mu;
            const float rstd = rsqrtf(var + EPSLN);
            if (tid < UU) {
                // pending rank-1 update for next step (applied lazily in B')
                sh_coef[tid] = LAMBC * (sh_cur[tid] - sh_hC[tid]);
                sh_hold[tid] = sh_cur[tid];
                const float nrm = (sh_y[tid] - mu) * rstd;
                sh_prev[tid] = tanhf(fmaf(nrm, lng[tid], lnb[tid]));
            }
        }
        __syncthreads();
    }

    // ---- Final dense head + softmax (tiny: 10 outputs) ----
    if (tid < OO) {
        float acc = db[tid];
        for (int u = 0; u < UU; ++u) acc = fmaf(sh_prev[u], dw[(size_t)u * OO + tid], acc);
        sp[tid] = acc;
    }
    __syncthreads();
    if (tid < OO) {
        float mx = sp[0];
        #pragma unroll
        for (int i = 1; i < OO; ++i) mx = fmaxf(mx, sp[i]);
        float s = 0.f;
        #pragma unroll
        for (int i = 0; i < OO; ++i) s += expf(sp[i] - mx);
        out[(size_t)b * OO + tid] = expf(sp[tid] - mx) / s;
    }
}

extern "C" void kernel_launch(void* const* d_in, const int* in_sizes, int n_in,
                              void* d_out, int out_size, void* d_ws, size_t ws_size,
                              hipStream_t stream) {
    const float* x    = (const float*)d_in[0];
    const float* kmat = (const float*)d_in[1];
    const float* rec  = (const float*)d_in[2];
    const float* bias = (const float*)d_in[3];
    const float* lng  = (const float*)d_in[4];
    const float* lnb  = (const float*)d_in[5];
    const float* dw   = (const float*)d_in[6];
    const float* db   = (const float*)d_in[7];
    float* out = (float*)d_out;

    // Workspace layout: xk [T,B,U] f32 (32 MiB) | C [B,U,U] f32 (64 MiB)
    float* xk = (float*)d_ws;
    float* C  = xk + (size_t)TT * BB * UU;

    const int waves = TT * (BB / 16) * (UU / 64);   // 8192 macro-tile waves
    xk_gemm_kernel<<<waves / 8, 256, 0, stream>>>(x, kmat, bias, xk);
    scan_kernel<<<BB, 1024, 0, stream>>>(xk, rec, lng, lnb, dw, db, C, out);
}